// EdgeLoss_87651692577051
// MI455X (gfx1250) — compile-verified
//
#include <hip/hip_runtime.h>
#include <stdint.h>

// ---------------------------------------------------------------------------
// EdgeLoss (class-balanced BCE-with-logits) for gfx1250 / MI455X.
// Single streaming pass (201 MB @ 23.3 TB/s ~ 8.6 us floor) using the CDNA5
// async global->LDS copy engine (non-temporal hints: data is single-use and
// larger than the 192 MB L2), double-buffered, then a scalar finalize.
// ---------------------------------------------------------------------------

#define BLOCK 256
#define TILE_ELEMS (BLOCK * 4)     // each lane moves one float4 per tensor
#define GRID 2048

// Issue a 16-byte async copy global -> LDS (tracked by ASYNCcnt).
// Non-temporal: streaming single-use data, don't churn WGP$/L2.
__device__ __forceinline__ void async_copy_b128(const float4* gsrc, uint32_t lds_off) {
    asm volatile("global_load_async_to_lds_b128 %0, %1, off th:TH_LOAD_NT"
                 :: "v"(lds_off), "v"(gsrc)
                 : "memory");
}

// Wait until ASYNCcnt <= N (per-wave counter; async loads complete in order).
#define WAIT_ASYNC(N) asm volatile("s_wait_asynccnt " #N ::: "memory")

// Low 32 bits of a flat shared-memory pointer == LDS byte offset (aperture in
// the high bits; HW adds the wave's LDS_BASE).
__device__ __forceinline__ uint32_t lds_off(const void* p) {
    return (uint32_t)(uintptr_t)p;
}

__device__ __forceinline__ void accum_elem(float xv, float tv,
                                           float& sp, float& sn,
                                           float& cp, float& cn) {
    float ax  = __builtin_fabsf(xv);
    float sf  = __logf(1.0f + __expf(-ax));              // log1p(exp(-|x|))
    float bce = fmaxf(xv, 0.0f) - xv * tv + sf;
    bool pos = (tv == 1.0f);
    bool neg = (tv == 0.0f);
    sp += pos ? bce : 0.0f;
    sn += neg ? bce : 0.0f;
    cp += pos ? 1.0f : 0.0f;
    cn += neg ? 1.0f : 0.0f;
}

__global__ void edge_loss_init(float* acc) {
    if (threadIdx.x < 4) acc[threadIdx.x] = 0.0f;
}

__global__ __launch_bounds__(BLOCK)
void edge_loss_main(const float* __restrict__ x, const float* __restrict__ t,
                    float* __restrict__ acc, long long n) {
    __shared__ float4 xin[2][BLOCK];
    __shared__ float4 tin[2][BLOCK];
    __shared__ float  red[4][BLOCK / 32];

    const int tid = threadIdx.x;
    const long long ntiles = n / TILE_ELEMS;

    float sp = 0.0f, sn = 0.0f, cp = 0.0f, cn = 0.0f;

    const uint32_t offx[2] = { lds_off(&xin[0][tid]), lds_off(&xin[1][tid]) };
    const uint32_t offt[2] = { lds_off(&tin[0][tid]), lds_off(&tin[1][tid]) };

    const float4* xg = (const float4*)x;
    const float4* tg = (const float4*)t;

    long long tile = blockIdx.x;
    if (tile < ntiles) {
        long long base = tile * (long long)BLOCK + tid;   // float4 index
        async_copy_b128(xg + base, offx[0]);
        async_copy_b128(tg + base, offt[0]);
    }

    int buf = 0;
    for (; tile < ntiles; tile += gridDim.x) {
        long long next = tile + gridDim.x;
        if (next < ntiles) {
            long long base = next * (long long)BLOCK + tid;
            async_copy_b128(xg + base, offx[buf ^ 1]);
            async_copy_b128(tg + base, offt[buf ^ 1]);
            WAIT_ASYNC(2);          // current tile's 2 copies done; next 2 in flight
        } else {
            WAIT_ASYNC(0);
        }
        __syncthreads();            // all waves' copies for this tile landed in LDS

        float4 xv = xin[buf][tid];  // ds_load_b128
        float4 tv = tin[buf][tid];
        accum_elem(xv.x, tv.x, sp, sn, cp, cn);
        accum_elem(xv.y, tv.y, sp, sn, cp, cn);
        accum_elem(xv.z, tv.z, sp, sn, cp, cn);
        accum_elem(xv.w, tv.w, sp, sn, cp, cn);

        __syncthreads();            // done reading buf before it is refilled
        buf ^= 1;
    }

    // Tail (n not a multiple of TILE_ELEMS): plain grid-stride scalar loads.
    for (long long i = ntiles * TILE_ELEMS + (long long)blockIdx.x * BLOCK + tid;
         i < n; i += (long long)gridDim.x * BLOCK) {
        float xv = __builtin_nontemporal_load(&x[i]);
        float tv = __builtin_nontemporal_load(&t[i]);
        accum_elem(xv, tv, sp, sn, cp, cn);
    }

    // wave32 tree reduction
    for (int o = 16; o > 0; o >>= 1) {
        sp += __shfl_down(sp, o);
        sn += __shfl_down(sn, o);
        cp += __shfl_down(cp, o);
        cn += __shfl_down(cn, o);
    }
    const int lane = tid & 31, wave = tid >> 5;
    if (lane == 0) {
        red[0][wave] = sp; red[1][wave] = sn;
        red[2][wave] = cp; red[3][wave] = cn;
    }
    __syncthreads();
    if (tid == 0) {
        float a = 0, b = 0, c = 0, d = 0;
        for (int w = 0; w < BLOCK / 32; ++w) {
            a += red[0][w]; b += red[1][w]; c += red[2][w]; d += red[3][w];
        }
        atomicAdd(&acc[0], a);
        atomicAdd(&acc[1], b);
        atomicAdd(&acc[2], c);
        atomicAdd(&acc[3], d);
    }
}

__global__ void edge_loss_finalize(const float* __restrict__ acc,
                                   float* __restrict__ out, long long n) {
    if (threadIdx.x == 0 && blockIdx.x == 0) {
        double spos = acc[0], sneg = acc[1];
        double cpos = acc[2], cneg = acc[3];
        double summ = cpos + cneg;
        double num  = cneg * spos + cpos * sneg;   // Sum(w * bce) * summ
        double res  = (summ > 0.0) ? num / (summ * (double)n) : 0.0;
        out[0] = (float)res;
    }
}

extern "C" void kernel_launch(void* const* d_in, const int* in_sizes, int n_in,
                              void* d_out, int out_size, void* d_ws, size_t ws_size,
                              hipStream_t stream) {
    const float* x   = (const float*)d_in[0];
    const float* tgt = (const float*)d_in[1];
    float*       out = (float*)d_out;
    float*       acc = (float*)d_ws;
    long long    n   = (long long)in_sizes[0];

    edge_loss_init<<<1, 32, 0, stream>>>(acc);
    edge_loss_main<<<GRID, BLOCK, 0, stream>>>(x, tgt, acc, n);
    edge_loss_finalize<<<1, 32, 0, stream>>>(acc, out, n);
}